// LaplacianPyramid_26474178413071
// MI455X (gfx1250) — compile-verified
//
#include <hip/hip_runtime.h>
#include <stdint.h>

// ---------------------------------------------------------------------------
// LaplacianPyramid grid_sample (border, align_corners=False) for MI455X.
//
// Roofline: 106 MB mandatory HBM traffic (~4.5us @ 23.3 TB/s) but the kernel
// is gather-issue bound: 134M divergent 4B gathers. All images (5.3 MB) fit in
// the 192 MB L2. Optimizations:
//  * Stage layer3 (256KB) + layer4 (64KB) = 320KB (exactly one WGP's LDS)
//    into LDS via the CDNA5 Tensor Data Mover (tensor_load_to_lds), so half
//    of all gathers become ds_load_b32 on the LDS pipe, concurrent with the
//    vmem pipe serving layer1/layer2 gathers out of L2.
//  * Grid reads / output writes are pure streams -> non-temporal (TH_NT) so
//    they don't evict the layer1/layer2 gather working set from WGP$/L2.
// ---------------------------------------------------------------------------

typedef unsigned int v4u __attribute__((ext_vector_type(4)));
typedef int          v8i __attribute__((ext_vector_type(8)));
typedef int          v4i __attribute__((ext_vector_type(4)));
typedef float        v2f __attribute__((ext_vector_type(2)));

#define L3_N 256
#define L4_N 128
#define L3_ELEMS (L3_N * L3_N)           // 65536 floats = 256 KB
#define L4_ELEMS (L4_N * L4_N)           // 16384 floats =  64 KB
#define SMEM_BYTES ((L3_ELEMS + L4_ELEMS) * sizeof(float))  // 320 KB

#define BLOCKS  512
#define THREADS 1024

// Issue one TDM descriptor copying a contiguous block of `dim0*dim1` floats
// (2-D tile: dim0 elements per row, dim1 rows, row stride == dim0) from
// global memory to LDS byte offset `lds_byte_off`.
// D# layout per CDNA5 ISA ch.8 (group0: count/lds_addr/global_addr/type,
// group1: data_size/dims/tile/strides). Groups 2/3 zero => short 2-group
// encoding (<=2D tensor, VADDR2/VADDR3 = NULL).
__device__ __forceinline__ void tdm_copy_to_lds(const float* src,
                                                unsigned lds_byte_off,
                                                unsigned dim0, unsigned dim1) {
  unsigned long long ga = (unsigned long long)(uintptr_t)src;
  v4u g0;
  g0[0] = 1u;                                   // count=1, is_restore=0, no gather
  g0[1] = lds_byte_off;                         // lds_addr (bytes)
  g0[2] = (unsigned)ga;                         // global_addr[31:0]
  g0[3] = ((unsigned)(ga >> 32) & 0x01FFFFFFu)  // global_addr[56:32]
          | 0x80000000u;                        // type=2 ("image")
  v8i g1;
  g1[0] = (int)(2u << 16);                               // data_size=2 (4B); mask=0
  g1[1] = (int)((dim0 & 0xFFFFu) << 16);                 // tensor_dim0[15:0]
  g1[2] = (int)((dim0 >> 16) | ((dim1 & 0xFFFFu) << 16)); // td0[31:16] | td1[15:0]
  g1[3] = (int)((dim1 >> 16) | ((dim0 & 0xFFFFu) << 16)); // td1[31:16] | tile_dim0
  g1[4] = (int)(dim1 & 0xFFFFu);                         // tile_dim1 | tile_dim2=0
  g1[5] = (int)dim0;                                     // tensor_dim0_stride[31:0]
  g1[6] = 0;                                             // stride0[47:32] | stride1 lo
  g1[7] = 0;
  v4i z4 = {0, 0, 0, 0};
  v8i z8 = {0, 0, 0, 0, 0, 0, 0, 0};
  __builtin_amdgcn_tensor_load_to_lds(g0, g1, z4, z4, z8, 0);
}

// Bilinear sample of an N x N image, border padding, align_corners=False.
// ix = clamp(((g+1)*N - 1)*0.5, 0, N-1) == clamp(fma(g, N/2, N/2 - 0.5), ...)
template <int N>
__device__ __forceinline__ float bilin(const float* __restrict__ img,
                                       float gx, float gy) {
  const float h = 0.5f * (float)N;
  const float b = h - 0.5f;
  float ix = fminf(fmaxf(fmaf(gx, h, b), 0.0f), (float)(N - 1));
  float iy = fminf(fmaxf(fmaf(gy, h, b), 0.0f), (float)(N - 1));
  int x0 = (int)ix;                 // ix >= 0, trunc == floor
  int y0 = (int)iy;
  float wx = ix - (float)x0;
  float wy = iy - (float)y0;
  int x1 = min(x0 + 1, N - 1);
  int y1 = min(y0 + 1, N - 1);
  const float* r0 = img + y0 * N;
  const float* r1 = img + y1 * N;
  float v00 = r0[x0], v01 = r0[x1];
  float v10 = r1[x0], v11 = r1[x1];
  float top = fmaf(wx, v01 - v00, v00);
  float bot = fmaf(wx, v11 - v10, v10);
  return fmaf(wy, bot - top, top);
}

__global__ __launch_bounds__(THREADS, 1)
void LaplacianPyramid_26474178413071_kernel(const v2f* __restrict__ grid,
                                            const float* __restrict__ l1,
                                            const float* __restrict__ l2,
                                            const float* __restrict__ l3,
                                            const float* __restrict__ l4,
                                            float* __restrict__ out, int n) {
  extern __shared__ float smem[];  // [0,256K): layer3 ; [256K,320K): layer4

  // Wave 0 issues the two TDM copies; everyone waits on TENSORcnt + barrier.
  if (threadIdx.x == 0) {
    tdm_copy_to_lds(l3, 0u, 16384u, 4u);                                   // 256 KB
    tdm_copy_to_lds(l4, (unsigned)(L3_ELEMS * sizeof(float)), 16384u, 1u); //  64 KB
  }
  __builtin_amdgcn_s_wait_tensorcnt(0);
  __syncthreads();

  const float* s3 = smem;
  const float* s4 = smem + L3_ELEMS;

  int tid = blockIdx.x * blockDim.x + threadIdx.x;
  int stride = gridDim.x * blockDim.x;      // 524288 threads -> 16 pts/thread

  #pragma unroll 2
  for (int p = tid; p < n; p += stride) {
    __builtin_prefetch(&grid[p + stride], 0, 1);  // global_prefetch_b8 (stream)
    v2f g = __builtin_nontemporal_load(&grid[p]);      // NT: don't pollute WGP$
    float acc = bilin<1024>(l1, g.x, g.y);   // global gathers (L2-resident)
    acc      += bilin< 512>(l2, g.x, g.y);   // global gathers (L2-resident)
    acc      += bilin< L3_N>(s3, g.x, g.y);  // LDS gathers (ds_load_b32)
    acc      += bilin< L4_N>(s4, g.x, g.y);  // LDS gathers (ds_load_b32)
    __builtin_nontemporal_store(acc, &out[p]);         // NT streaming store
  }
}

extern "C" void kernel_launch(void* const* d_in, const int* in_sizes, int n_in,
                              void* d_out, int out_size, void* d_ws, size_t ws_size,
                              hipStream_t stream) {
  const v2f*   grid = (const v2f*)d_in[0];      // (B,Ho,Wo,2) f32
  const float* l1   = (const float*)d_in[1];    // 1024x1024
  const float* l2   = (const float*)d_in[2];    // 512x512
  const float* l3   = (const float*)d_in[3];    // 256x256
  const float* l4   = (const float*)d_in[4];    // 128x128
  float* out = (float*)d_out;                   // (B,1,Ho,Wo) f32

  LaplacianPyramid_26474178413071_kernel<<<dim3(BLOCKS), dim3(THREADS),
                                           SMEM_BYTES, stream>>>(
      grid, l1, l2, l3, l4, out, out_size);
}